// HMU_49520972923446
// MI455X (gfx1250) — compile-verified
//
#include <hip/hip_runtime.h>
#include <math.h>

typedef __attribute__((ext_vector_type(2))) float v2f;
typedef __attribute__((ext_vector_type(4))) float v4f;
typedef __attribute__((ext_vector_type(8))) float v8f;

#define DCOLS   1024            // feature dim D
#define KRANK   8               // rank of v
#define CB      32              // columns of x staged per chunk
#define NCHUNK  (DCOLS / CB)    // 32
#define XSTRIDE 36              // padded LDS row stride (banks 4*(9r mod 16)+c -> conflict free)
#define WAVES   8               // waves per block
#define ROWS_PER_BLOCK (WAVES * 16)

// quad[b] = lam*||x_b-mu||^2 + sum_k (v_k . (x_b-mu))^2 ; out = exp(-quad/D)
//
// WMMA trick: B columns 0..7 hold v_0..v_7, columns 8..15 hold the SAME rows
// again (no masking). C columns 8..15 then duplicate columns 0..7, so the
// 16-lane sum-of-squares reduction gives exactly 2*sum_k p_k^2; we scale by
// 0.5 in the epilogue (bit-exact). This removes all per-WMMA cndmask VALU.
__global__ __launch_bounds__(256) void
hess_quad_exp_kernel(const float* __restrict__ x, const float* __restrict__ mu,
                     const float* __restrict__ lamp, const float* __restrict__ vmat,
                     float* __restrict__ out)
{
    // v in paired-transposed layout: lds_v[(d>>1)*16 + n*2 + (d&1)] = v[n][d]
    __shared__ __align__(16) float lds_v[(DCOLS / 2) * 16];        // 32 KB
    __shared__ __align__(16) float lds_x[WAVES * 16 * XSTRIDE];    // 18 KB

    const int tid  = threadIdx.x;
    const int lane = tid & 31;
    const int wave = tid >> 5;
    const int h    = lane >> 4;   // half-wave: selects K sub-pair {2h, 2h+1}
    const int n16  = lane & 15;   // A-row (M) / B-col (N) index

    // ---- one-time cooperative stage of v into LDS (coalesced global reads)
    for (int idx = tid; idx < KRANK * DCOLS; idx += 256) {
        int n = idx >> 10;            // 0..7
        int d = idx & (DCOLS - 1);
        lds_v[((d >> 1) << 4) + (n << 1) + (d & 1)] = vmat[idx];
    }
    __syncthreads();

    const float lam = lamp[0];
    const long  rowbase  = (long)blockIdx.x * ROWS_PER_BLOCK + wave * 16;
    const float* xrowbase = x + rowbase * DCOLS;
    float* tile = &lds_x[wave * 16 * XSTRIDE];   // wave-private tile, no barriers needed

    const int r4 = lane >> 3;     // staging row subgroup 0..3
    const int c4 = lane & 7;      // staging float4 column 0..7

    v8f acc  = {0.f, 0.f, 0.f, 0.f, 0.f, 0.f, 0.f, 0.f};
    v2f nrm2 = {0.f, 0.f};        // packed ||diff||^2 partial -> v_pk_fma_f32

    // ---- software-pipelined prefetch of chunk 0 (streaming, non-temporal)
    v4f mur = *(const v4f*)&mu[c4 * 4];
    v4f xr0 = __builtin_nontemporal_load((const v4f*)&xrowbase[( 0 + r4) * DCOLS + c4 * 4]);
    v4f xr1 = __builtin_nontemporal_load((const v4f*)&xrowbase[( 4 + r4) * DCOLS + c4 * 4]);
    v4f xr2 = __builtin_nontemporal_load((const v4f*)&xrowbase[( 8 + r4) * DCOLS + c4 * 4]);
    v4f xr3 = __builtin_nontemporal_load((const v4f*)&xrowbase[(12 + r4) * DCOLS + c4 * 4]);

    for (int c = 0; c < NCHUNK; ++c) {
        // store diff = x - mu for this chunk into the wave's LDS tile
        *(v4f*)&tile[( 0 + r4) * XSTRIDE + c4 * 4] = xr0 - mur;
        *(v4f*)&tile[( 4 + r4) * XSTRIDE + c4 * 4] = xr1 - mur;
        *(v4f*)&tile[( 8 + r4) * XSTRIDE + c4 * 4] = xr2 - mur;
        *(v4f*)&tile[(12 + r4) * XSTRIDE + c4 * 4] = xr3 - mur;

        // prefetch chunk c+1 (overlaps with the WMMA work below)
        if (c + 1 < NCHUNK) {
            const int cb = (c + 1) * CB;
            mur = *(const v4f*)&mu[cb + c4 * 4];
            xr0 = __builtin_nontemporal_load((const v4f*)&xrowbase[( 0 + r4) * DCOLS + cb + c4 * 4]);
            xr1 = __builtin_nontemporal_load((const v4f*)&xrowbase[( 4 + r4) * DCOLS + cb + c4 * 4]);
            xr2 = __builtin_nontemporal_load((const v4f*)&xrowbase[( 8 + r4) * DCOLS + cb + c4 * 4]);
            xr3 = __builtin_nontemporal_load((const v4f*)&xrowbase[(12 + r4) * DCOLS + cb + c4 * 4]);
        }

        // A fragment: lane holds diff[row n16][d + 2h .. +1]
        // B fragment: v[(n16&7)][d + 2h .. +1]  (columns 8..15 duplicate 0..7 on purpose)
        const float* arow = &tile[n16 * XSTRIDE + 2 * h];
        const float* brow = &lds_v[((c * CB + 2 * h) >> 1) * 16 + (n16 & 7) * 2];
#pragma unroll
        for (int kk = 0; kk < CB / 4; ++kk) {
            v2f a = *(const v2f*)&arow[kk * 4];
            v2f b = *(const v2f*)&brow[kk * 32];
            nrm2 += a * a;                       // v_pk_fma_f32 on VALU, co-issues with WMMA
            acc = __builtin_amdgcn_wmma_f32_16x16x4_f32(
                false, a, false, b, (short)0, acc, false, false);
        }
    }

    // ---- epilogue
    // full norm of row n16: combine packed halves, then the two half-wave partials
    float nrm = nrm2.x + nrm2.y;
    nrm += __shfl_xor(nrm, 16, 32);

    // acc[r] on lane L = p[M = r + 8*(L>>4)][N = L&15]; sum of squares over all 16 N
    // equals 2 * sum_{k<8} p_k^2 because columns 8..15 duplicate 0..7.
    float red[8];
#pragma unroll
    for (int r = 0; r < 8; ++r) {
        float s = acc[r] * acc[r];
        s += __shfl_xor(s, 1, 32);
        s += __shfl_xor(s, 2, 32);
        s += __shfl_xor(s, 4, 32);
        s += __shfl_xor(s, 8, 32);
        red[r] = s;   // identical across each 16-lane half
    }

    // lane L (L<8 or L>=24) owns row M = L&15 ; its (doubled) sumsq is red[L&7]
    const int rsel = lane & 7;
    float ssq = red[0];
#pragma unroll
    for (int r = 1; r < 8; ++r) ssq = (rsel == r) ? red[r] : ssq;

    const float quad = fmaf(lam, nrm, 0.5f * ssq);
    const float res  = expf(quad * (-1.0f / (float)DCOLS));

    if ((lane < 8) || (lane >= 24)) {
        out[rowbase + n16] = res;
    }
}

extern "C" void kernel_launch(void* const* d_in, const int* in_sizes, int n_in,
                              void* d_out, int out_size, void* d_ws, size_t ws_size,
                              hipStream_t stream) {
    const float* x    = (const float*)d_in[0];
    const float* mu   = (const float*)d_in[1];
    const float* lamp = (const float*)d_in[2];
    const float* vmat = (const float*)d_in[3];
    float* out = (float*)d_out;

    const int Dsz = in_sizes[1];            // 1024
    const int B   = in_sizes[0] / Dsz;      // 131072
    const int nblocks = B / ROWS_PER_BLOCK; // 1024 (B is a multiple of 128)

    hess_quad_exp_kernel<<<nblocks, 256, 0, stream>>>(x, mu, lamp, vmat, out);
}